// OneLayerGCNWithGlobalAdg_17824114279162
// MI455X (gfx1250) — compile-verified
//
#include <hip/hip_runtime.h>

typedef __attribute__((ext_vector_type(2))) float v2f;
typedef __attribute__((ext_vector_type(8))) float v8f;

#define NN     131072
#define EE     2097152
#define DIN    512
#define DOUT   64
#define BB     256

#define WMMA4(a, b, c) __builtin_amdgcn_wmma_f32_16x16x4_f32(false, (a), false, (b), (short)0, (c), false, false)

// ---------------------------------------------------------------------------
// Init helpers
// ---------------------------------------------------------------------------
__global__ void zero_i32(int* __restrict__ p, int n) {
    int i = blockIdx.x * blockDim.x + threadIdx.x;
    if (i < n) p[i] = 0;
}
__global__ void zero_f32k(float* __restrict__ p, int n) {
    int i = blockIdx.x * blockDim.x + threadIdx.x;
    if (i < n) p[i] = 0.0f;
}

// ---------------------------------------------------------------------------
// Wt[64,512] = W[512,64]^T  (so B fragments along K are contiguous b64 loads)
// ---------------------------------------------------------------------------
__global__ void transpose_w(const float* __restrict__ Wm, float* __restrict__ Wt) {
    int i = blockIdx.x * blockDim.x + threadIdx.x;     // i in [0, 512*64)
    if (i >= DIN * DOUT) return;
    int k = i >> 6, n = i & 63;
    Wt[(size_t)n * DIN + k] = Wm[i];
}

// ---------------------------------------------------------------------------
// C[M,64] = A[M,512] x W using V_WMMA_F32_16X16X4_F32.
// One wave -> 32(M) x 64(N): 2 A-frags x 4 B-frags = 8 v8f accumulators.
// Per k-step: 2+4 global_load_b64, 8 WMMAs.
// act!=0: fuse bias + PReLU (anchor branch).
// ---------------------------------------------------------------------------
__global__ __launch_bounds__(256) void gemm_wmma_f32(
        const float* __restrict__ A, const float* __restrict__ Wt,
        const float* __restrict__ bias, const float* __restrict__ alpha,
        float* __restrict__ C, int M, int act) {
    const int lane  = threadIdx.x & 31;
    const int wave  = threadIdx.x >> 5;
    const int mbase = (blockIdx.x * 8 + wave) * 32;    // 32 rows per wave
    if (mbase >= M) return;

    const int half  = lane >> 4;                       // 0 or 1
    const int lidx  = lane & 15;
    const int khalf = half * 2;                        // K offset in step: 0 or 2
    const float* __restrict__ ar0 = A + (size_t)(mbase + lidx) * DIN;
    const float* __restrict__ ar1 = A + (size_t)(mbase + 16 + lidx) * DIN;
    const float* __restrict__ bc  = Wt + (size_t)lidx * DIN;

    v8f c00 = {}, c01 = {}, c02 = {}, c03 = {};
    v8f c10 = {}, c11 = {}, c12 = {}, c13 = {};

    for (int k = 0; k < DIN; k += 4) {
        const int kk = k + khalf;
        v2f a0 = *(const v2f*)(ar0 + kk);
        v2f a1 = *(const v2f*)(ar1 + kk);
        v2f b0 = *(const v2f*)(bc + kk);
        v2f b1 = *(const v2f*)(bc + 16 * DIN + kk);
        v2f b2 = *(const v2f*)(bc + 32 * DIN + kk);
        v2f b3 = *(const v2f*)(bc + 48 * DIN + kk);
        c00 = WMMA4(a0, b0, c00);  c01 = WMMA4(a0, b1, c01);
        c02 = WMMA4(a0, b2, c02);  c03 = WMMA4(a0, b3, c03);
        c10 = WMMA4(a1, b0, c10);  c11 = WMMA4(a1, b1, c11);
        c12 = WMMA4(a1, b2, c12);  c13 = WMMA4(a1, b3, c13);
    }

    const float al = alpha[0];
    #pragma unroll
    for (int f = 0; f < 2; ++f) {
        const int mrow = mbase + f * 16 + half * 8;
        #pragma unroll
        for (int r = 0; r < 8; ++r) {
            const size_t mo = (size_t)(mrow + r) * DOUT;
            float v0 = f ? c10[r] : c00[r];
            float v1 = f ? c11[r] : c01[r];
            float v2 = f ? c12[r] : c02[r];
            float v3 = f ? c13[r] : c03[r];
            if (act) {
                v0 += bias[lidx];      v0 = (v0 >= 0.f) ? v0 : al * v0;
                v1 += bias[16 + lidx]; v1 = (v1 >= 0.f) ? v1 : al * v1;
                v2 += bias[32 + lidx]; v2 = (v2 >= 0.f) ? v2 : al * v2;
                v3 += bias[48 + lidx]; v3 = (v3 >= 0.f) ? v3 : al * v3;
            }
            C[mo + lidx]      = v0;
            C[mo + 16 + lidx] = v1;
            C[mo + 32 + lidx] = v2;
            C[mo + 48 + lidx] = v3;
        }
    }
}

// ---------------------------------------------------------------------------
// CSR-by-dst construction: count, scan, fill.
// ---------------------------------------------------------------------------
__global__ void count_deg(const int* __restrict__ dst, int* __restrict__ deg) {
    int e = blockIdx.x * blockDim.x + threadIdx.x;
    if (e >= EE) return;
    __hip_atomic_fetch_add(&deg[dst[e]], 1, __ATOMIC_RELAXED, __HIP_MEMORY_SCOPE_AGENT);
}

// per-block exclusive scan of 256 elements (Hillis-Steele in LDS)
__global__ void scan_block(const int* __restrict__ deg, int* __restrict__ base,
                           int* __restrict__ bsums) {
    __shared__ int sh[256];
    const int tid = threadIdx.x;
    const int gid = blockIdx.x * 256 + tid;
    const int v = deg[gid];
    sh[tid] = v;
    __syncthreads();
    #pragma unroll
    for (int off = 1; off < 256; off <<= 1) {
        int t = (tid >= off) ? sh[tid - off] : 0;
        __syncthreads();
        if (tid >= off) sh[tid] += t;
        __syncthreads();
    }
    base[gid] = sh[tid] - v;                 // exclusive
    if (tid == 255) bsums[blockIdx.x] = sh[255];
}

// single-block exclusive scan of the 512 block sums
__global__ void scan_tops(int* __restrict__ bsums) {
    __shared__ int sh[512];
    const int tid = threadIdx.x;             // blockDim = 512
    const int v = bsums[tid];
    sh[tid] = v;
    __syncthreads();
    #pragma unroll
    for (int off = 1; off < 512; off <<= 1) {
        int t = (tid >= off) ? sh[tid - off] : 0;
        __syncthreads();
        if (tid >= off) sh[tid] += t;
        __syncthreads();
    }
    bsums[tid] = sh[tid] - v;                // exclusive
}

__global__ void scan_add(int* __restrict__ base, const int* __restrict__ bsums) {
    int gid = blockIdx.x * 256 + threadIdx.x;
    base[gid] += bsums[blockIdx.x];
    if (gid == 0) base[NN] = EE;
}

// scatter {src, w} pairs into dst buckets (one b64 store per edge)
__global__ void fill_buckets(const int* __restrict__ src, const int* __restrict__ dst,
                             const float* __restrict__ ew, const int* __restrict__ base,
                             int* __restrict__ cursor, int2* __restrict__ bucket) {
    int e = blockIdx.x * blockDim.x + threadIdx.x;
    if (e >= EE) return;
    const int d = dst[e];
    const int p = base[d] + __hip_atomic_fetch_add(&cursor[d], 1,
                        __ATOMIC_RELAXED, __HIP_MEMORY_SCOPE_AGENT);
    int2 t;
    t.x = src[e];
    t.y = __float_as_int(ew[e]);
    bucket[p] = t;
}

// ---------------------------------------------------------------------------
// Gather aggregation (no f32 atomics on the N*64 stream):
// one wave per dst node; lane owns 2 channels; fuse bias+PReLU+pooling.
// h0 (32MB) is L2-resident, so the random per-edge row gathers hit L2.
// ---------------------------------------------------------------------------
__global__ __launch_bounds__(256) void aggregate(
        const float* __restrict__ h0, const int2* __restrict__ bucket,
        const int* __restrict__ base, const float* __restrict__ bias,
        const float* __restrict__ alpha, const int* __restrict__ seg,
        float* __restrict__ h_out, float* __restrict__ pool_sum,
        float* __restrict__ cnt) {
    const long long gid = (long long)blockIdx.x * blockDim.x + threadIdx.x;
    const int node = (int)(gid >> 5);
    if (node >= NN) return;
    const int lane = (int)(gid & 31);
    const int c    = lane * 2;

    const int b0 = base[node];
    const int b1 = base[node + 1];
    float a0 = 0.f, a1 = 0.f;
    for (int i = b0; i < b1; ++i) {
        const int2 t = bucket[i];                       // broadcast (uniform) load
        const float w = __int_as_float(t.y);
        const v2f hv = *(const v2f*)(h0 + (size_t)t.x * DOUT + c);
        a0 = fmaf(w, hv.x, a0);
        a1 = fmaf(w, hv.y, a1);
    }

    const float al = alpha[0];
    float v0 = a0 + bias[c];     v0 = (v0 >= 0.f) ? v0 : al * v0;
    float v1 = a1 + bias[c + 1]; v1 = (v1 >= 0.f) ? v1 : al * v1;

    v2f outv; outv.x = v0; outv.y = v1;
    *(v2f*)(h_out + (size_t)node * DOUT + c) = outv;

    const int s = seg[node];
    __hip_atomic_fetch_add(&pool_sum[(size_t)s * DOUT + c],     v0,
                           __ATOMIC_RELAXED, __HIP_MEMORY_SCOPE_AGENT);
    __hip_atomic_fetch_add(&pool_sum[(size_t)s * DOUT + c + 1], v1,
                           __ATOMIC_RELAXED, __HIP_MEMORY_SCOPE_AGENT);
    if (lane == 0)
        __hip_atomic_fetch_add(&cnt[s], 1.0f,
                               __ATOMIC_RELAXED, __HIP_MEMORY_SCOPE_AGENT);
}

// ---------------------------------------------------------------------------
// Pool divide: mean = sum / max(count, 1)
// ---------------------------------------------------------------------------
__global__ void pool_div(const float* __restrict__ pool_sum,
                         const float* __restrict__ cnt,
                         float* __restrict__ outp) {
    const int gid = blockIdx.x * blockDim.x + threadIdx.x;
    if (gid >= BB * DOUT) return;
    float c = cnt[gid >> 6];
    c = (c > 1.0f) ? c : 1.0f;
    outp[gid] = pool_sum[gid] / c;
}

extern "C" void kernel_launch(void* const* d_in, const int* in_sizes, int n_in,
                              void* d_out, int out_size, void* d_ws, size_t ws_size,
                              hipStream_t stream) {
    const float* in_feat  = (const float*)d_in[0];
    const float* anchor   = (const float*)d_in[1];
    const float* edge_w   = (const float*)d_in[2];
    const float* Wm       = (const float*)d_in[3];
    const float* bias     = (const float*)d_in[4];
    const float* alpha    = (const float*)d_in[5];
    const int*   edge_src = (const int*)d_in[6];
    const int*   edge_dst = (const int*)d_in[7];
    const int*   seg_ids  = (const int*)d_in[8];

    float* out      = (float*)d_out;
    float* h_out    = out;                                    // [N,64]
    float* pool_out = out + (size_t)NN * DOUT;                // [B,64]
    float* anch_out = out + (size_t)NN * DOUT + BB * DOUT;    // [B,64]

    // workspace layout (all sub-buffers 256B aligned)
    char* ws = (char*)d_ws;
    float* h0      = (float*)ws;                      ws += (size_t)NN * DOUT * 4;   // 32MB
    int2*  bucket  = (int2*)ws;                       ws += (size_t)EE * 8;          // 16MB
    int*   base    = (int*)ws;                        ws += ((size_t)NN + 64) * 4;
    int*   deg     = (int*)ws;                        ws += (size_t)NN * 4;
    int*   cursor  = (int*)ws;                        ws += (size_t)NN * 4;          // deg+cursor contiguous
    int*   bsums   = (int*)ws;                        ws += 512 * 4;
    float* Wt      = (float*)ws;                      ws += (size_t)DIN * DOUT * 4;  // 128KB
    float* pool_sum= (float*)ws;                      ws += (size_t)BB * DOUT * 4;
    float* cnt     = (float*)ws;                                                    // pool_sum+cnt contiguous

    // 1) zero deg+cursor (contiguous 2N ints) and pool_sum+cnt
    zero_i32<<<(2 * NN + 255) / 256, 256, 0, stream>>>(deg, 2 * NN);
    zero_f32k<<<(BB * DOUT + BB + 255) / 256, 256, 0, stream>>>(pool_sum, BB * DOUT + BB);

    // 2) transpose W for contiguous-K B fragments
    transpose_w<<<(DIN * DOUT + 255) / 256, 256, 0, stream>>>(Wm, Wt);

    // 3) node GEMM h0 = in_feat @ W   (512 blocks x 8 waves x 32 rows)
    gemm_wmma_f32<<<(NN + 255) / 256, 256, 0, stream>>>(
        in_feat, Wt, bias, alpha, h0, NN, 0);

    // 4) anchor GEMM anchor_out = PReLU(anchor @ W + bias)
    gemm_wmma_f32<<<(BB + 255) / 256, 256, 0, stream>>>(
        anchor, Wt, bias, alpha, anch_out, BB, 1);

    // 5) CSR-by-dst build: count -> scan -> fill
    count_deg<<<EE / 256, 256, 0, stream>>>(edge_dst, deg);
    scan_block<<<NN / 256, 256, 0, stream>>>(deg, base, bsums);
    scan_tops<<<1, 512, 0, stream>>>(bsums);
    scan_add<<<NN / 256, 256, 0, stream>>>(base, bsums);
    fill_buckets<<<EE / 256, 256, 0, stream>>>(edge_src, edge_dst, edge_w,
                                               base, cursor, bucket);

    // 6) gather aggregation + fused bias/PReLU + pooling partial sums
    aggregate<<<(NN * 32) / 256, 256, 0, stream>>>(
        h0, bucket, base, bias, alpha, seg_ids, h_out, pool_sum, cnt);

    // 7) pooled mean
    pool_div<<<(BB * DOUT + 255) / 256, 256, 0, stream>>>(pool_sum, cnt, pool_out);
}